// RoiPoolingConv_85272280694976
// MI455X (gfx1250) — compile-verified
//
#include <hip/hip_runtime.h>
#include <hip/hip_bf16.h>
#include <stdint.h>

// ROI bilinear resize (16x16 -> 7x7, 512 ch, 16 ROIs, batch 16) on gfx1250.
// Memory-bound: ~160MB traffic -> ~7us at 23.3 TB/s. Interpolation done as two
// separable GEMM passes on the f32 WMMA pipe (V_WMMA_F32_16X16X4_F32), with the
// crop staged in LDS via gfx1250 async global->LDS copies (ASYNCcnt tracked).

typedef __attribute__((ext_vector_type(2))) float v2f;
typedef __attribute__((ext_vector_type(8))) float v8f;

// Builtin-probed types: global_load_async_to_lds_b128 takes
//   (int4-vector __device__(AS1)* src, int4-vector __shared__(AS3)* dst, imm offset, imm cpol)
typedef int v4i_ __attribute__((vector_size(16)));
typedef __attribute__((address_space(1))) v4i_ gv4i;
typedef __attribute__((address_space(3))) v4i_ lv4i;

#define CT     64               // channels per block
#define YSTR   (16 * CT + 16)   // padded crop row stride (floats) -> bank-conflict free
#define NW     4                // waves per block (128 threads, wave32)
#define TMPSZ  (7 * 16 * 16)    // per-wave intermediate [p][x][c16]

// jax.image.resize bilinear, antialias=False, 16 -> 7:
//   src(p) = (p + 0.5) * 16/7 - 0.5 = (32p + 9) / 14
// Weight of input row k for output row p (rows p>=7 are zero padding).
__device__ __forceinline__ float wval(int p, int k) {
    if (p >= 7) return 0.0f;
    int   num = 32 * p + 9;
    int   y0  = num / 14;
    float f   = (float)(num - y0 * 14) * (1.0f / 14.0f);
    if (k == y0)     return 1.0f - f;
    if (k == y0 + 1) return f;
    return 0.0f;
}

__global__ __launch_bounds__(128) void roi_resize_wmma(
    const float* __restrict__ img, float* __restrict__ out)
{
    __shared__ __align__(16) float crop[16 * YSTR];   // [y][x][c] padded
    __shared__ __align__(16) float tmp[NW * TMPSZ];   // per-wave [p][x][c16]

    const int tid  = threadIdx.x;
    const int lane = tid & 31;
    const int wave = tid >> 5;
    const int n    = lane & 15;   // matrix column (N) / A-matrix row (M)
    const int hi   = lane >> 4;   // half-wave selector
    const int klo  = hi * 2;      // K-pair base within a K=4 chunk

    const int ct  = blockIdx.x;          // channel tile (0..7)
    const int roi = blockIdx.y;          // 0..15
    const int b   = blockIdx.z;          // 0..15
    const int c0  = ct * CT;
    const int x0  = (roi & 3) * 16;      // ROI grid: x = 16*(roi%4)
    const int y0  = (roi >> 2) * 16;     //           y = 16*(roi/4)

    // ---------------- stage 16x16xCT crop into LDS (async to LDS) ----------------
    const float* gbase = img + (((size_t)b * 64 + y0) * 64 + x0) * 512 + c0;

#if __has_builtin(__builtin_amdgcn_global_load_async_to_lds_b128)
    #pragma unroll
    for (int i = 0; i < 32; ++i) {
        int q  = i * 128 + tid;
        int c4 = (q & 15) * 4;           // float4 within channel tile
        int x  = (q >> 4) & 15;
        int y  = q >> 8;
        __builtin_amdgcn_global_load_async_to_lds_b128(
            (gv4i*)(gbase + ((size_t)y * 64 + x) * 512 + c4),
            (lv4i*)&crop[y * YSTR + x * CT + c4], 0, 0);
    }
#if __has_builtin(__builtin_amdgcn_s_wait_asynccnt)
    __builtin_amdgcn_s_wait_asynccnt(0);
#else
    asm volatile("s_wait_asynccnt 0" ::: "memory");
#endif
#else
    #pragma unroll
    for (int i = 0; i < 32; ++i) {
        int q  = i * 128 + tid;
        int c4 = (q & 15) * 4;
        int x  = (q >> 4) & 15;
        int y  = q >> 8;
        const float4 v = *(const float4*)(gbase + ((size_t)y * 64 + x) * 512 + c4);
        *(float4*)&crop[y * YSTR + x * CT + c4] = v;
    }
#endif
    __syncthreads();

    // A fragments: weight matrix W (same for vertical and horizontal pass).
    // f32 16x4 A layout: lanes 0-15 hold M=lane, K={0,1}; lanes 16-31 K={2,3}.
    v2f Af[4];
    #pragma unroll
    for (int kk = 0; kk < 4; ++kk) {
        Af[kk].x = wval(n, kk * 4 + klo);
        Af[kk].y = wval(n, kk * 4 + klo + 1);
    }

    float*    tw = &tmp[wave * TMPSZ];
    const int cw = wave * 16;            // this wave's 16-channel slice

    // ---------------- pass 1: vertical 16 -> 7 (per x column) ----------------
    // Tmp[p][x][c] = sum_y W[p][y] * crop[y][x][c]; K=16 as 4 chained K=4 WMMAs.
    #pragma unroll
    for (int x = 0; x < 16; ++x) {
        v8f acc = {};
        #pragma unroll
        for (int kk = 0; kk < 4; ++kk) {
            int y = kk * 4 + klo;
            v2f bf;
            bf.x = crop[y * YSTR + x * CT + cw + n];
            bf.y = crop[(y + 1) * YSTR + x * CT + cw + n];
            acc = __builtin_amdgcn_wmma_f32_16x16x4_f32(
                false, Af[kk], false, bf, (short)0, acc, false, false);
        }
        // D layout: lane 0-15 VGPR v -> (M=v, N=lane); only p=0..6 are real.
        if (hi == 0) {
            #pragma unroll
            for (int p = 0; p < 7; ++p)
                tw[(p * 16 + x) * 16 + n] = acc[p];
        }
    }

    // ---------------- pass 2: horizontal 16 -> 7 (per output row) ----------------
    // Out[py][q][c] = sum_x W[q][x] * Tmp[py][x][c]
    float*    outp = out + (((size_t)b * 16 + roi) * 49) * 512;
    const int c    = c0 + cw + n;
    #pragma unroll
    for (int py = 0; py < 7; ++py) {
        v8f acc = {};
        #pragma unroll
        for (int kk = 0; kk < 4; ++kk) {
            int xk = kk * 4 + klo;
            v2f bf;
            bf.x = tw[(py * 16 + xk) * 16 + n];
            bf.y = tw[(py * 16 + xk + 1) * 16 + n];
            acc = __builtin_amdgcn_wmma_f32_16x16x4_f32(
                false, Af[kk], false, bf, (short)0, acc, false, false);
        }
        if (hi == 0) {
            #pragma unroll
            for (int q = 0; q < 7; ++q)
                outp[((size_t)py * 7 + q) * 512 + c] = acc[q];
        }
    }
}

extern "C" void kernel_launch(void* const* d_in, const int* in_sizes, int n_in,
                              void* d_out, int out_size, void* d_ws, size_t ws_size,
                              hipStream_t stream) {
    (void)in_sizes; (void)n_in; (void)out_size; (void)d_ws; (void)ws_size;
    const float* img = (const float*)d_in[0];
    float*       out = (float*)d_out;
    // grid: 8 channel tiles x 16 ROIs x 16 batches; 128 threads = 4 wave32
    dim3 grid(512 / CT, 16, 16);
    roi_resize_wmma<<<grid, dim3(128), 0, stream>>>(img, out);
}